// AuxLoss_4183298147157
// MI455X (gfx1250) — compile-verified
//
#include <hip/hip_runtime.h>
#include <math.h>

typedef float v2f __attribute__((ext_vector_type(2)));
typedef float v8f __attribute__((ext_vector_type(8)));

#define WAVES_PER_BLOCK 8
#define ROWS_PER_WAVE   32
#define ROWS_PER_BLOCK  (WAVES_PER_BLOCK * ROWS_PER_WAVE)   // 256 rows, 256 threads
#define NITEMS          48                                   // 3 stats x 16 classes
#define D_DIM           64
#define C_DIM           10

// f(u) = 0.5 - 0.5*erf((-0.5-u)/(sigma*sqrt2)), sigma=0.5  ==  0.5 + 0.5*erf((u+0.5)*sqrt2)
__device__ __forceinline__ float reg_f(float u) {
    return fmaf(0.5f, erff((u + 0.5f) * 1.41421356237309504f), 0.5f);
}

__global__ __launch_bounds__(ROWS_PER_BLOCK)
void auxloss_main(const int* __restrict__ yhat,
                  const float* __restrict__ yg,
                  const float* __restrict__ uzg,
                  float* __restrict__ partial,
                  int nRows, int nChunks)
{
    // Per-wave broadcast tables for WMMA operand construction.
    // valbuf[w][r] = {reg, ce, count, 0}; clsbuf[w][r] = class of row r.
    __shared__ float valbuf[WAVES_PER_BLOCK][ROWS_PER_WAVE][4];
    __shared__ int   clsbuf[WAVES_PER_BLOCK][ROWS_PER_WAVE];
    __shared__ float bseg[3][16];

    const int lane = threadIdx.x & 31;
    const int w    = threadIdx.x >> 5;
    const int tl   = lane & 15;      // N / M half-index for WMMA layouts
    const int h    = lane >> 4;      // lane half
    const int bidx = (tl < 3) ? tl : 3;   // which stat this lane's B column carries

    valbuf[w][lane][3] = 0.f;        // padding stat, never rewritten (per-wave region)

    // Persistent per-wave segment accumulator: D[m=class][n: 0=reg 1=ce 2=count]
    v8f acc = {0.f, 0.f, 0.f, 0.f, 0.f, 0.f, 0.f, 0.f};

    for (int chunk = blockIdx.x; chunk < nChunks; chunk += gridDim.x) {
        const long rowBase = (long)(chunk * WAVES_PER_BLOCK + w) * ROWS_PER_WAVE;
        const float* ub = uzg + rowBase * D_DIM;

        // ---- regularizer row sums: 32 rows x 64 cols, 2 rows per iteration ----
        // lane t reads float4 at flat offset c*128 + t*4 -> lies fully in row 2c + h
        #pragma unroll 4
        for (int c = 0; c < 16; ++c) {
            const long r = rowBase + 2 * c + h;
            float4 q;
            if (r < nRows) q = *(const float4*)(ub + c * 128 + lane * 4);
            else           q = make_float4(0.f, 0.f, 0.f, 0.f);
            float s = reg_f(q.x) + reg_f(q.y) + reg_f(q.z) + reg_f(q.w);
            s += __shfl_xor(s, 1, 32);
            s += __shfl_xor(s, 2, 32);
            s += __shfl_xor(s, 4, 32);
            s += __shfl_xor(s, 8, 32);
            if (tl == 0) valbuf[w][2 * c + h][0] = (r < nRows) ? s : 0.f;
        }

        // ---- per-row cross entropy: lane t handles row rowBase + t ----
        const long row  = rowBase + lane;
        float myce  = 0.f;
        float mycnt = 0.f;
        int   mycls = 0;
        if (row < nRows) {
            const float* yr = yg + row * C_DIM;
            float2 p0 = *(const float2*)(yr + 0);
            float2 p1 = *(const float2*)(yr + 2);
            float2 p2 = *(const float2*)(yr + 4);
            float2 p3 = *(const float2*)(yr + 6);
            float2 p4 = *(const float2*)(yr + 8);
            float vv[C_DIM] = {p0.x, p0.y, p1.x, p1.y, p2.x,
                               p2.y, p3.x, p3.y, p4.x, p4.y};
            float m = vv[0];
            #pragma unroll
            for (int j = 1; j < C_DIM; ++j) m = fmaxf(m, vv[j]);
            float se = 0.f;
            #pragma unroll
            for (int j = 0; j < C_DIM; ++j) se += expf(vv[j] - m);
            mycls = yhat[row];
            float tgt = vv[0];
            #pragma unroll
            for (int j = 1; j < C_DIM; ++j) tgt = (mycls == j) ? vv[j] : tgt;
            myce  = m + logf(se) - tgt;
            mycnt = 1.f;
        }
        valbuf[w][lane][1] = myce;
        valbuf[w][lane][2] = mycnt;
        clsbuf[w][lane]    = mycls;

        // ---- WMMA segment-sum: D += onehot(cls)[16xK] x vals[Kx16], K=4 rows ----
        // A lane layout (16x4 f32): vgpr j = (M = tl, K = 2h + j)
        // B lane layout (4x16 f32): vgpr j = (K = 2h + j, N = tl)
        // B element (k,n) = stat[n] of row (4g+k): one broadcast LDS read per element.
        #pragma unroll
        for (int g = 0; g < 8; ++g) {
            const int r0 = 4 * g + 2 * h;
            const int r1 = r0 + 1;
            v2f a, b;
            b.x = valbuf[w][r0][bidx];
            b.y = valbuf[w][r1][bidx];
            a.x = (clsbuf[w][r0] == tl) ? 1.f : 0.f;
            a.y = (clsbuf[w][r1] == tl) ? 1.f : 0.f;
            acc = __builtin_amdgcn_wmma_f32_16x16x4_f32(
                      /*neg_a=*/false, a, /*neg_b=*/false, b,
                      /*c_mod=*/(short)0, acc, /*reuse_a=*/false, /*reuse_b=*/false);
        }
    }

    // ---- block-level reduction of per-wave accumulators ----
    __syncthreads();
    if (threadIdx.x < NITEMS) ((float*)bseg)[threadIdx.x] = 0.f;
    __syncthreads();
    // D layout: lane t, vgpr v holds (M = v + 8h, N = tl)
    if (tl < 3) {
        #pragma unroll
        for (int v = 0; v < 8; ++v) {
            atomicAdd(&bseg[tl][v + 8 * h], acc[v]);
        }
    }
    __syncthreads();
    if (threadIdx.x < NITEMS) {
        partial[(size_t)threadIdx.x * gridDim.x + blockIdx.x] =
            ((float*)bseg)[threadIdx.x];   // item = stat*16 + class
    }
}

__global__ __launch_bounds__(1024)
void auxloss_finalize(const float* __restrict__ partial,
                      const float* __restrict__ lmbd,
                      float* __restrict__ out,
                      int nBlocks)
{
    __shared__ float seg[NITEMS];
    const int lane = threadIdx.x & 31;
    const int w    = threadIdx.x >> 5;          // 32 waves
    for (int item = w; item < NITEMS; item += 32) {
        float s = 0.f;
        for (int b = lane; b < nBlocks; b += 32)
            s += partial[(size_t)item * nBlocks + b];
        s += __shfl_xor(s, 1, 32);
        s += __shfl_xor(s, 2, 32);
        s += __shfl_xor(s, 4, 32);
        s += __shfl_xor(s, 8, 32);
        s += __shfl_xor(s, 16, 32);
        if (lane == 0) seg[item] = s;
    }
    __syncthreads();
    if (threadIdx.x == 0) {
        float regl = 0.f, auxl = 0.f, nuniq = 0.f;
        for (int c = 0; c < 16; ++c) {
            float cnt = seg[2 * 16 + c];
            if (cnt > 0.f) {
                nuniq += 1.f;
                regl  += seg[0 * 16 + c] / (cnt * (float)D_DIM);
                auxl  += seg[1 * 16 + c] / cnt;
            }
        }
        out[0] = auxl / nuniq + lmbd[0] * (regl / nuniq);
    }
}

extern "C" void kernel_launch(void* const* d_in, const int* in_sizes, int n_in,
                              void* d_out, int out_size, void* d_ws, size_t ws_size,
                              hipStream_t stream) {
    const int*   yhat = (const int*)d_in[0];
    const float* yg   = (const float*)d_in[1];
    const float* uzg  = (const float*)d_in[2];
    const float* lmbd = (const float*)d_in[3];
    const int nRows   = in_sizes[0];
    const int nChunks = (nRows + ROWS_PER_BLOCK - 1) / ROWS_PER_BLOCK;

    int nBlocks = 2048;
    while (nBlocks > 1 && (size_t)NITEMS * nBlocks * sizeof(float) > ws_size)
        nBlocks >>= 1;
    if (nBlocks > nChunks) nBlocks = nChunks;
    if (nBlocks < 1) nBlocks = 1;

    float* partial = (float*)d_ws;
    auxloss_main<<<nBlocks, ROWS_PER_BLOCK, 0, stream>>>(yhat, yg, uzg, partial,
                                                         nRows, nChunks);
    auxloss_finalize<<<1, 1024, 0, stream>>>(partial, lmbd, (float*)d_out, nBlocks);
}